// SelectiveNeighborhoodAttention_22093311771269
// MI455X (gfx1250) — compile-verified
//
#include <hip/hip_runtime.h>
#include <math.h>

// ---- problem constants (from reference) ----
#define DIM      128
#define HEADS    4
#define HEAD_DIM 32      // == wave32
#define KS       7
#define HH       64
#define WW       64
#define BB       4
#define NPIX     (BB*HH*WW)   // 16384 rows for all GEMMs

// neighborhood-union tiling for the WMMA attention kernel
#define UW       22            // union window width for 16 consecutive pixels
#define NKEY     (KS*UW)       // 154 real keys
#define NKEYP    160           // padded to 10 WMMA n-tiles
#define KVSTRIDE 140           // LDS row stride (floats) -> bank-conflict-free
#define PSTRIDE  164           // P row stride (floats)   -> bank-conflict-free

typedef __attribute__((ext_vector_type(2))) float v2f;
typedef __attribute__((ext_vector_type(8))) float v8f;

// -------------------------------------------------------------------------
// Generic f32 WMMA GEMM: C[M,N] = A[M,K] @ W[K,N] + bias, one 16x16 tile per
// wave32 using V_WMMA_F32_16X16X4_F32 (full fp32 matrix pipe).
// mode 0: scale columns < DIM by HEAD_DIM^-0.5 (q-scaling for qkv GEMM)
// mode 1: ReLU (gate hidden layer)
// Weights <=192KB -> fully L2-resident, no LDS staging needed at this size.
// -------------------------------------------------------------------------
__global__ __launch_bounds__(256) void gemm_wmma_f32(
    const float* __restrict__ A, const float* __restrict__ W,
    const float* __restrict__ bias, float* __restrict__ C,
    int M, int N, int K, int mode)
{
    const int lane = threadIdx.x & 31;
    const int wave = threadIdx.x >> 5;
    const int ntn  = N >> 4;
    const int tile = blockIdx.x * 8 + wave;
    const int tm   = tile / ntn;
    const int tn   = tile - tm * ntn;
    if (tm * 16 >= M) return;              // wave-uniform guard (EXEC stays all-1)

    const int l16 = lane & 15;
    const int hi  = lane >> 4;             // 0: K pair (0,1); 1: K pair (2,3)
    const int row = tm * 16 + l16;         // A-matrix M = lane%16
    const int col = tn * 16 + l16;         // B-matrix N = lane%16

    v8f acc = {};
    const float* arow = A + (size_t)row * K + 2 * hi;
    const float* wcol = W + (size_t)(2 * hi) * N + col;

    for (int k0 = 0; k0 < K; k0 += 4) {
        v2f a = *(const v2f*)(arow + k0);          // global_load_b64, 8B aligned
        v2f b;
        b.x = wcol[(size_t)k0 * N];
        b.y = wcol[(size_t)(k0 + 1) * N];
        acc = __builtin_amdgcn_wmma_f32_16x16x4_f32(
                  false, a, false, b, (short)0, acc, false, false);
    }

    #pragma unroll
    for (int r = 0; r < 8; ++r) {
        int m = tm * 16 + r + 8 * hi;              // C/D layout: VGPR r -> row r (+8 hi)
        float v = acc[r] + bias[col];
        if (mode == 0) { if (col < DIM) v *= 0.17677669529663687f; } // 1/sqrt(32)
        else           { v = fmaxf(v, 0.0f); }
        C[(size_t)m * N + col] = v;
    }
}

// -------------------------------------------------------------------------
// WMMA neighborhood attention.
// Block = 128 threads = 4 waves, one wave per head; block covers 16 pixels
// of one image row. All 16 pixels share 7 key-rows, and their 7-wide column
// windows fall inside a fixed 22-column union -> 154 keys (padded to 160).
//   S(16x160) = Q(16x32) @ Kt(32x160)      : 80 f32 WMMAs
//   mask (-inf) + rpb bias + row softmax   : 4-step half-wave shuffles
//   O(16x32)  = P(16x160) @ V(160x32)      : 80 f32 WMMAs
// K then V staged through one LDS buffer ([key][dim], stride 140 floats ->
// conflict-free for both B-operand access patterns). P goes through LDS
// (stride 164, conflict-free) to convert C-layout -> A-operand layout.
// qkv (25MB) is fully L2-resident, so staging loads hit L2.
// -------------------------------------------------------------------------
__global__ __launch_bounds__(128) void natten_wmma(
    const float* __restrict__ qkv, const float* __restrict__ rpb,
    float* __restrict__ attn_out)
{
    extern __shared__ float smem[];
    float* kv  = smem;                               // [160][140]  K, then V
    float* pb  = smem + NKEYP * KVSTRIDE;            // [4][16][164] P per head
    float* rb  = pb + HEADS * 16 * PSTRIDE;          // [4*169] rpb copy
    int*   nix = (int*)(rb + HEADS * 169);           // [160] key -> pixel idx

    const int tid  = threadIdx.x;
    const int lane = tid & 31;
    const int h    = tid >> 5;                       // wave id == head
    const int l16  = lane & 15;
    const int hi   = lane >> 4;

    const int gx = blockIdx.x;                       // 4 xt * 64 y * 4 b
    const int x0 = (gx & 3) << 4;
    const int y  = (gx >> 2) & 63;
    const int b  = gx >> 8;

    const int sh  = min(max(y - KS / 2, 0), HH - KS);
    const int cu0 = min(max(x0 - KS / 2, 0), WW - UW);

    // ---- stage rpb, key->pixel indices (pad keys clamp to key 153) ----
    for (int i = tid; i < HEADS * 169; i += 128) rb[i] = rpb[i];
    for (int key = tid; key < NKEYP; key += 128) {
        int kk = min(key, NKEY - 1);
        int ki = kk / UW, kj = kk - ki * UW;
        nix[key] = (b * HH + sh + ki) * WW + (cu0 + kj);
    }
    __syncthreads();

    // ---- stage K[key][0:128] (coalesced 512B per wave-instruction) ----
    for (int idx = tid; idx < NKEYP * 32; idx += 128) {     // 40 iters, uniform
        int key = idx >> 5;
        int c4  = (idx & 31) << 2;
        float4 t = *(const float4*)(qkv + (size_t)nix[key] * 384 + 128 + c4);
        *(float4*)(kv + key * KVSTRIDE + c4) = t;
    }
    __syncthreads();

    // ---- QK^T via WMMA: A = Q (global), B = Kt (LDS) ----
    const int pixbase = (b * HH + y) * WW + x0;
    const float* qbase = qkv + (size_t)(pixbase + l16) * 384 + h * 32 + 2 * hi;

    const v8f vzero = {};
    v8f sc[10];
    #pragma unroll
    for (int t = 0; t < 10; ++t) sc[t] = vzero;

    for (int k0 = 0; k0 < HEAD_DIM; k0 += 4) {
        v2f a = *(const v2f*)(qbase + k0);           // Q[m=l16][k0+2hi, +1]
        #pragma unroll
        for (int t = 0; t < 10; ++t) {
            v2f bk = *(const v2f*)(kv + (t * 16 + l16) * KVSTRIDE + k0 + 2 * hi);
            sc[t] = __builtin_amdgcn_wmma_f32_16x16x4_f32(
                        false, a, false, bk, (short)0, sc[t], false, false);
        }
    }
    __syncthreads();     // all waves done reading K -> buffer reusable

    // ---- stage V into the same LDS buffer ----
    for (int idx = tid; idx < NKEYP * 32; idx += 128) {
        int key = idx >> 5;
        int c4  = (idx & 31) << 2;
        float4 t = *(const float4*)(qkv + (size_t)nix[key] * 384 + 256 + c4);
        *(float4*)(kv + key * KVSTRIDE + c4) = t;
    }

    // ---- mask + bias + row softmax (rows live in C-layout: m = r + 8*hi) ----
    const int bh0 = sh - y + KS - 1;
    float* ph = pb + h * 16 * PSTRIDE;
    float rmax[8], rsum[8];
    #pragma unroll
    for (int r = 0; r < 8; ++r) rmax[r] = -3.0e38f;

    #pragma unroll
    for (int t = 0; t < 10; ++t) {
        const int key   = t * 16 + l16;
        const int ki    = key / UW;
        const int kj    = key - ki * UW;
        const int kx    = cu0 + kj;
        const bool keyok = key < NKEY;
        #pragma unroll
        for (int r = 0; r < 8; ++r) {
            const int m   = r + 8 * hi;
            const int xm  = x0 + m;
            const int swm = min(max(xm - KS / 2, 0), WW - KS);
            float s = -3.0e38f;
            if (keyok && kx >= swm && kx < swm + KS)
                s = sc[t][r] + rb[h * 169 + (ki + bh0) * 13 + (kx - xm + KS - 1)];
            sc[t][r] = s;
            rmax[r] = fmaxf(rmax[r], s);
        }
    }
    #pragma unroll
    for (int r = 0; r < 8; ++r) {        // masks < 16 keep each half-wave intact
        float v = rmax[r];
        v = fmaxf(v, __shfl_xor(v, 1, 32));
        v = fmaxf(v, __shfl_xor(v, 2, 32));
        v = fmaxf(v, __shfl_xor(v, 4, 32));
        v = fmaxf(v, __shfl_xor(v, 8, 32));
        rmax[r] = v;
        rsum[r] = 0.0f;
    }
    #pragma unroll
    for (int t = 0; t < 10; ++t) {
        #pragma unroll
        for (int r = 0; r < 8; ++r) {
            float p = __expf(sc[t][r] - rmax[r]);    // masked/pad -> exactly 0
            rsum[r] += p;
            ph[(r + 8 * hi) * PSTRIDE + t * 16 + l16] = p;
        }
    }
    #pragma unroll
    for (int r = 0; r < 8; ++r) {
        float v = rsum[r];
        v += __shfl_xor(v, 1, 32);
        v += __shfl_xor(v, 2, 32);
        v += __shfl_xor(v, 4, 32);
        v += __shfl_xor(v, 8, 32);
        rsum[r] = v;
    }
    __syncthreads();     // V staged + P visible to all lanes

    // ---- P @ V via WMMA: A = P (LDS), B = V (LDS) ----
    const float* pa = ph + l16 * PSTRIDE + 2 * hi;
    #pragma unroll
    for (int nt = 0; nt < 2; ++nt) {
        v8f acc = vzero;
        for (int k0 = 0; k0 < NKEYP; k0 += 4) {
            v2f a = *(const v2f*)(pa + k0);          // P[m=l16][k0+2hi, +1]
            v2f bv;
            bv.x = kv[(k0 + 2 * hi    ) * KVSTRIDE + h * 32 + nt * 16 + l16];
            bv.y = kv[(k0 + 2 * hi + 1) * KVSTRIDE + h * 32 + nt * 16 + l16];
            acc = __builtin_amdgcn_wmma_f32_16x16x4_f32(
                      false, a, false, bv, (short)0, acc, false, false);
        }
        #pragma unroll
        for (int r = 0; r < 8; ++r) {
            int m = r + 8 * hi;
            attn_out[(size_t)(pixbase + m) * DIM + h * 32 + nt * 16 + l16] =
                acc[r] / rsum[r];
        }
    }
}

// -------------------------------------------------------------------------
// gate[m] = sigmoid(dot(relu_hidden[m, 0:64], g2_w) + g2_b)
// -------------------------------------------------------------------------
__global__ void gate_kernel(const float* __restrict__ g1h,
                            const float* __restrict__ g2w,
                            const float* __restrict__ g2b,
                            float* __restrict__ gate, int M)
{
    const int m = blockIdx.x * blockDim.x + threadIdx.x;
    if (m >= M) return;
    float acc = g2b[0];
    const float* row = g1h + (size_t)m * 64;
    #pragma unroll 8
    for (int i = 0; i < 64; ++i) acc = fmaf(row[i], g2w[i], acc);
    gate[m] = 1.0f / (1.0f + __expf(-acc));
}

// -------------------------------------------------------------------------
// Final: out = (attn @ proj_w + proj_b) * gate + x   (WMMA f32, fused)
// -------------------------------------------------------------------------
__global__ __launch_bounds__(256) void proj_final_wmma(
    const float* __restrict__ A, const float* __restrict__ Wp,
    const float* __restrict__ bias, const float* __restrict__ gate,
    const float* __restrict__ x, float* __restrict__ out)
{
    const int lane = threadIdx.x & 31;
    const int wave = threadIdx.x >> 5;
    const int tile = blockIdx.x * 8 + wave;
    const int tm   = tile >> 3;            // N/16 = 8 column tiles
    const int tn   = tile & 7;
    const int l16  = lane & 15;
    const int hi   = lane >> 4;
    const int row  = tm * 16 + l16;
    const int col  = tn * 16 + l16;

    v8f acc = {};
    const float* arow = A  + (size_t)row * DIM + 2 * hi;
    const float* wcol = Wp + (size_t)(2 * hi) * DIM + col;

    for (int k0 = 0; k0 < DIM; k0 += 4) {
        v2f a = *(const v2f*)(arow + k0);
        v2f b;
        b.x = wcol[(size_t)k0 * DIM];
        b.y = wcol[(size_t)(k0 + 1) * DIM];
        acc = __builtin_amdgcn_wmma_f32_16x16x4_f32(
                  false, a, false, b, (short)0, acc, false, false);
    }

    #pragma unroll
    for (int r = 0; r < 8; ++r) {
        const int m = tm * 16 + r + 8 * hi;
        const float g = gate[m];
        const size_t idx = (size_t)m * DIM + col;
        out[idx] = (acc[r] + bias[col]) * g + x[idx];
    }
}

// -------------------------------------------------------------------------
extern "C" void kernel_launch(void* const* d_in, const int* in_sizes, int n_in,
                              void* d_out, int out_size, void* d_ws, size_t ws_size,
                              hipStream_t stream)
{
    const float* x      = (const float*)d_in[0];
    const float* qkv_w  = (const float*)d_in[1];
    const float* qkv_b  = (const float*)d_in[2];
    const float* proj_w = (const float*)d_in[3];
    const float* proj_b = (const float*)d_in[4];
    const float* rpb    = (const float*)d_in[5];
    const float* g1_w   = (const float*)d_in[6];
    const float* g1_b   = (const float*)d_in[7];
    const float* g2_w   = (const float*)d_in[8];
    const float* g2_b   = (const float*)d_in[9];
    float* out = (float*)d_out;

    // workspace layout (floats): qkv | g1 hidden | gate | attn  (~38 MB)
    float* ws   = (float*)d_ws;
    float* qkv  = ws;                                  // 16384*384
    float* g1h  = qkv  + (size_t)NPIX * 384;           // 16384*64
    float* gate = g1h  + (size_t)NPIX * 64;            // 16384
    float* attn = gate + NPIX;                         // 16384*128

    // 1) qkv = x @ qkv_w + qkv_b (q columns pre-scaled): 1024*24 tiles / 8 waves
    gemm_wmma_f32<<<3072, 256, 0, stream>>>(x, qkv_w, qkv_b, qkv,
                                            NPIX, 3 * DIM, DIM, 0);
    // 2) hidden = relu(x @ g1_w + g1_b): 1024*4 tiles / 8 waves
    gemm_wmma_f32<<<512, 256, 0, stream>>>(x, g1_w, g1_b, g1h,
                                           NPIX, DIM / 2, DIM, 1);
    // 3) gate = sigmoid(hidden @ g2_w + g2_b)
    gate_kernel<<<NPIX / 256, 256, 0, stream>>>(g1h, g2_w, g2_b, gate, NPIX);
    // 4) WMMA neighborhood attention: one block per (b, y, 16-pixel segment)
    {
        const size_t smem_bytes =
            (size_t)(NKEYP * KVSTRIDE + HEADS * 16 * PSTRIDE + HEADS * 169 + NKEYP)
            * sizeof(float);                            // ~135 KB -> 2 blocks/WGP
        natten_wmma<<<BB * HH * (WW / 16), 128, smem_bytes, stream>>>(qkv, rpb, attn);
    }
    // 5) out = (attn @ proj_w + proj_b) * gate + x: 1024*8 tiles / 8 waves
    proj_final_wmma<<<1024, 256, 0, stream>>>(attn, proj_w, proj_b, gate, x, out);
}